// GraphAttentionLayer_10874857193700
// MI455X (gfx1250) — compile-verified
//
#include <hip/hip_runtime.h>
#include <math.h>

typedef __attribute__((ext_vector_type(2))) float v2f;
typedef __attribute__((ext_vector_type(8))) float v8f;

#define DIMS 128
#define TILE 16

__device__ __forceinline__ float gelu_tanh(float x) {
    // tanh-approx gelu (jax.nn.gelu approximate=True / tfa gelu)
    const float c0 = 0.7978845608028654f;  // sqrt(2/pi)
    const float c1 = 0.044715f;
    float x3 = x * x * x;
    float t = tanhf(c0 * (x + c1 * x3));
    return 0.5f * x * (1.0f + t);
}

// C[row, col] = gelu(A[row,:] @ W[:,col] + b[col])
// A: [nrows,128], W: [128,128], b: [128], C: [nrows,128]
// grid.x = nrows/16 row tiles; block = 256 threads = 8 waves; wave w owns
// column tile w (8 * 16 = 128 columns). Uses native f32 WMMA 16x16x4.
__global__ __launch_bounds__(256)
void proj_gelu_wmma_kernel(const float* __restrict__ A,
                           const float* __restrict__ W,
                           const float* __restrict__ b,
                           float* __restrict__ C) {
    const int wave = threadIdx.x >> 5;
    const int lane = threadIdx.x & 31;
    const int half = lane >> 4;       // which 16-lane group
    const int l    = lane & 15;
    const int row0 = blockIdx.x * TILE;
    const int col0 = wave * TILE;

    v8f acc = {};
    const float* arow = A + (size_t)(row0 + l) * DIMS;

    #pragma unroll
    for (int k0 = 0; k0 < DIMS; k0 += 4) {
        // A frag (16x4 f32): lane 16h+l holds A[row0+l][k0+2h], A[row0+l][k0+2h+1]
        v2f af = *(const v2f*)(arow + k0 + 2 * half);
        // B frag (4x16 f32): lane 16h+l holds W[k0+2h][col0+l], W[k0+2h+1][col0+l]
        v2f bf;
        bf.x = W[(size_t)(k0 + 2 * half + 0) * DIMS + col0 + l];
        bf.y = W[(size_t)(k0 + 2 * half + 1) * DIMS + col0 + l];
        acc = __builtin_amdgcn_wmma_f32_16x16x4_f32(
            /*neg_a=*/false, af, /*neg_b=*/false, bf,
            /*c_mod=*/(short)0, acc, /*reuse_a=*/false, /*reuse_b=*/false);
    }

    const float bias = b[col0 + l];
    #pragma unroll
    for (int j = 0; j < 8; ++j) {
        // C frag: VGPR j at lane 16h+l = C[row0 + j + 8h][col0 + l]
        const int row = row0 + j + 8 * half;
        C[(size_t)row * DIMS + col0 + l] = gelu_tanh(acc[j] + bias);
    }
}

__global__ void zero_f32_kernel(float* __restrict__ p, long long n) {
    long long i = (long long)blockIdx.x * blockDim.x + threadIdx.x;
    if (i < n) p[i] = 0.0f;
}

// One wave per edge: gather q[r], k[c] (float4/lane), wave-reduce dot,
// sigmoid, then atomic scatter-add coef * v[c] into out[r].
__global__ __launch_bounds__(256)
void edge_aggregate_kernel(const float* __restrict__ q,
                           const float* __restrict__ k,
                           const float* __restrict__ v,
                           const int* __restrict__ erows,
                           const int* __restrict__ ecols,
                           float* __restrict__ out,
                           int nedges) {
    const int wavesPerBlock = blockDim.x >> 5;
    const int e = blockIdx.x * wavesPerBlock + (threadIdx.x >> 5);
    if (e >= nedges) return;
    const int lane = threadIdx.x & 31;

    const int r = erows[e];
    const int c = ecols[e];

    const float4 qv = ((const float4*)(q + (size_t)r * DIMS))[lane];
    const float4 kv = ((const float4*)(k + (size_t)c * DIMS))[lane];
    float dot = qv.x * kv.x + qv.y * kv.y + qv.z * kv.z + qv.w * kv.w;

    #pragma unroll
    for (int off = 16; off >= 1; off >>= 1)
        dot += __shfl_xor(dot, off, 32);

    const float scale = 0.08838834764831845f;  // 1/sqrt(128)
    const float coef = 1.0f / (1.0f + __expf(-dot * scale));

    const float4 vv = ((const float4*)(v + (size_t)c * DIMS))[lane];
    float* orow = out + (size_t)r * DIMS + lane * 4;
    __hip_atomic_fetch_add(orow + 0, coef * vv.x, __ATOMIC_RELAXED, __HIP_MEMORY_SCOPE_AGENT);
    __hip_atomic_fetch_add(orow + 1, coef * vv.y, __ATOMIC_RELAXED, __HIP_MEMORY_SCOPE_AGENT);
    __hip_atomic_fetch_add(orow + 2, coef * vv.z, __ATOMIC_RELAXED, __HIP_MEMORY_SCOPE_AGENT);
    __hip_atomic_fetch_add(orow + 3, coef * vv.w, __ATOMIC_RELAXED, __HIP_MEMORY_SCOPE_AGENT);
}

extern "C" void kernel_launch(void* const* d_in, const int* in_sizes, int n_in,
                              void* d_out, int out_size, void* d_ws, size_t ws_size,
                              hipStream_t stream) {
    const float* query = (const float*)d_in[0];
    const float* memry = (const float*)d_in[1];
    const float* Wq    = (const float*)d_in[2];
    const float* bq    = (const float*)d_in[3];
    const float* Wk    = (const float*)d_in[4];
    const float* bk    = (const float*)d_in[5];
    const float* Wv    = (const float*)d_in[6];
    const float* bv    = (const float*)d_in[7];
    const int* erows   = (const int*)d_in[8];
    const int* ecols   = (const int*)d_in[9];
    float* out         = (float*)d_out;

    const int N = in_sizes[0] / DIMS;   // 100000 query rows
    const int M = in_sizes[1] / DIMS;   // 100000 memory rows
    const int E = in_sizes[8];          // 1600000 edges

    // Workspace layout: q | k | v (all f32 [.,128])
    float* qbuf = (float*)d_ws;
    float* kbuf = qbuf + (size_t)N * DIMS;
    float* vbuf = kbuf + (size_t)M * DIMS;

    // Projections with gelu on the f32 WMMA path (N, M are multiples of 16).
    proj_gelu_wmma_kernel<<<N / TILE, 256, 0, stream>>>(query, Wq, bq, qbuf);
    proj_gelu_wmma_kernel<<<M / TILE, 256, 0, stream>>>(memry, Wk, bk, kbuf);
    proj_gelu_wmma_kernel<<<M / TILE, 256, 0, stream>>>(memry, Wv, bv, vbuf);

    // Zero the output (harness poisons it) before atomic segment-sum.
    const long long outN = (long long)N * DIMS;
    zero_f32_kernel<<<(int)((outN + 255) / 256), 256, 0, stream>>>(out, outN);

    // Edge-wise attention + scatter-add (one wave per edge, 8 edges/block).
    const int edgeBlocks = (E + 7) / 8;
    edge_aggregate_kernel<<<edgeBlocks, 256, 0, stream>>>(qbuf, kbuf, vbuf,
                                                          erows, ecols, out, E);
}